// CRFCell_62852551410110
// MI455X (gfx1250) — compile-verified
//
#include <hip/hip_runtime.h>

typedef __attribute__((ext_vector_type(2))) float v2f;
typedef __attribute__((ext_vector_type(8))) float v8f;

constexpr int CB = 256;  // batch
constexpr int CS = 512;  // sequence length
constexpr int CT = 64;   // num tags
constexpr int LDP = 66;  // padded LDS row stride (floats)

// ---------------------------------------------------------------------------
// Denominator: forward algorithm, 16 batches per wave32 block.
// Exact log-space step via f32 WMMA:
//   nxt[b,j] = m_b + log( sum_i exp(score[b,i]-m_b) * exp(trans[i,j]) ) + em[b,s,j]
// B operand (exp(trans)) is loop-invariant -> held in 128 VGPRs for all 511
// steps; A loaded once per step and reused across the 4 output j-tiles.
// ---------------------------------------------------------------------------
__global__ __launch_bounds__(32) void crf_den_kernel(
    const float* __restrict__ em, const int* __restrict__ mask,
    const float* __restrict__ trans, const float* __restrict__ start_t,
    const float* __restrict__ end_t, float* __restrict__ den)
{
  __shared__ __align__(16) float sco[16][LDP];   // running scores
  __shared__ __align__(16) float P[16][LDP];     // exp(score - m)
  __shared__ float pm[32];
  __shared__ float ps[32];
  __shared__ float mrow[16];
  __shared__ int   mskAll[16 * CS];              // mask tile, preloaded once

  const int lane = (int)threadIdx.x;
  const int b0   = (int)blockIdx.x * 16;
  const int msel = lane >> 4;   // half-wave select (ISA K striping)
  const int nlo  = lane & 15;

  // mask tile -> LDS (coalesced, once)
  for (int idx = lane; idx < 16 * CS; idx += 32)
    mskAll[idx] = mask[b0 * CS + idx];

  // init: score[b,j] = start_t[j] + em[b,0,j]
  for (int idx = lane; idx < 16 * CT; idx += 32) {
    int b = idx >> 6, j = idx & 63;
    sco[b][j] = start_t[j] + em[(size_t)(b0 + b) * CS * CT + j];
  }

  // Loop-invariant B operands in registers.
  // ISA 4x16 f32 B striping: lanes 0-15 hold (K+0,K+1), lanes 16-31 (K+2,K+3),
  // N = lane&15. Bv[n][kc] covers K-chunk kc*4, output columns n*16..n*16+15.
  v2f Bv[4][16];
#pragma unroll
  for (int n = 0; n < 4; ++n)
#pragma unroll
    for (int kc = 0; kc < 16; ++kc) {
      const int kb = kc * 4 + (msel << 1);
      Bv[n][kc].x = __expf(trans[(kb + 0) * CT + n * 16 + nlo]);
      Bv[n][kc].y = __expf(trans[(kb + 1) * CT + n * 16 + nlo]);
    }

  // Emission base in WMMA C/D layout: reg r, tile n ->
  //   batch b0 + r + 8*msel, column j = 16n + nlo
  const float* embase = em + ((size_t)(b0 + msel * 8) * CS) * CT + nlo;
  __syncthreads();

  for (int s = 1; s < CS; ++s) {
    // Issue this step's emission loads early; plenty of compute before use.
    float emc[8][4];
#pragma unroll
    for (int r = 0; r < 8; ++r)
#pragma unroll
      for (int n = 0; n < 4; ++n)
        emc[r][n] = embase[((size_t)r * CS + s) * CT + n * 16];
    {
      const int spf = (s + 8 < CS) ? s + 8 : CS - 1;
#pragma unroll
      for (int r = 0; r < 8; ++r)
        __builtin_prefetch(embase + ((size_t)r * CS + spf) * CT, 0, 0);
    }

    // Fused row-max + exp (2 lanes per batch, scores kept in registers).
    float scf[32];
    {
      const int b = lane >> 1, h = lane & 1;
      float mx = -3.0e38f;
#pragma unroll
      for (int j = 0; j < 32; ++j) {
        scf[j] = sco[b][h * 32 + j];
        mx = fmaxf(mx, scf[j]);
      }
      pm[lane] = mx;
    }
    __syncthreads();
    {
      const int b = lane >> 1, h = lane & 1;
      const float m = fmaxf(pm[2 * b], pm[2 * b + 1]);
      if (h == 0) mrow[b] = m;
#pragma unroll
      for (int j = 0; j < 32; ++j)
        P[b][h * 32 + j] = __expf(scf[j] - m);
    }
    __syncthreads();

    // A operands once per step (ISA 16x4 f32 A striping), reused for all n.
    v2f A[16];
#pragma unroll
    for (int kc = 0; kc < 16; ++kc)
      A[kc] = *(const v2f*)&P[nlo][kc * 4 + (msel << 1)];

    float mreg[8];
    int   mkreg[8];
#pragma unroll
    for (int r = 0; r < 8; ++r) {
      mreg[r]  = mrow[r + msel * 8];
      mkreg[r] = mskAll[(r + msel * 8) * CS + s];
    }

    // GEMM: 4 independent accumulator chains, all operands in registers.
    v8f acc[4] = {};
#pragma unroll
    for (int kc = 0; kc < 16; ++kc)
#pragma unroll
      for (int n = 0; n < 4; ++n)
        acc[n] = __builtin_amdgcn_wmma_f32_16x16x4_f32(
            false, A[kc], false, Bv[n][kc], (short)0, acc[n], false, false);

    // nxt = m_b + log(acc) + em ; masked score update.
#pragma unroll
    for (int n = 0; n < 4; ++n)
#pragma unroll
      for (int r = 0; r < 8; ++r) {
        const int bb = r + msel * 8;
        const float nx  = mreg[r] + __logf(acc[n][r]) + emc[r][n];
        const float old = sco[bb][n * 16 + nlo];
        sco[bb][n * 16 + nlo] = mkreg[r] ? nx : old;
      }
    __syncthreads();
  }

  // den[b] = logsumexp_j(score[b,j] + end_t[j])
  for (int idx = lane; idx < 16 * CT; idx += 32) {
    int b = idx >> 6, j = idx & 63;
    sco[b][j] += end_t[j];
  }
  __syncthreads();
  {
    const int b = lane >> 1, h = lane & 1;
    float mx = -3.0e38f;
    for (int j = h * 32; j < h * 32 + 32; ++j) mx = fmaxf(mx, sco[b][j]);
    pm[lane] = mx;
  }
  __syncthreads();
  {
    const int b = lane >> 1, h = lane & 1;
    const float m = fmaxf(pm[2 * b], pm[2 * b + 1]);
    float ssum = 0.f;
    for (int j = h * 32; j < h * 32 + 32; ++j) ssum += __expf(sco[b][j] - m);
    ps[lane] = ssum;
  }
  __syncthreads();
  if (lane < 16) {
    const float m = fmaxf(pm[2 * lane], pm[2 * lane + 1]);
    den[b0 + lane] = m + __logf(ps[2 * lane] + ps[2 * lane + 1]);
  }
}

// ---------------------------------------------------------------------------
// Numerator: gold-path score per batch; deterministic fixed-order reduction.
// ---------------------------------------------------------------------------
__global__ __launch_bounds__(64) void crf_num_kernel(
    const float* __restrict__ em, const int* __restrict__ mask,
    const int* __restrict__ labels, const float* __restrict__ trans,
    const float* __restrict__ start_t, const float* __restrict__ end_t,
    float* __restrict__ num)
{
  const int b = (int)blockIdx.x;
  const int t = (int)threadIdx.x;
  __shared__ float part[64];
  __shared__ int   lenp[64];

  float acc = 0.f;
  int cnt = 0;
  for (int s = t; s < CS; s += 64) {
    cnt += (mask[b * CS + s] != 0) ? 1 : 0;
    if (s >= 1) {
      const float mk = (float)mask[b * CS + s];
      const int prev = labels[b * CS + s - 1];
      const int cur  = labels[b * CS + s];
      acc += mk * (trans[prev * CT + cur] +
                   em[((size_t)b * CS + s) * CT + cur]);
    }
  }
  part[t] = acc;
  lenp[t] = cnt;
  __syncthreads();
  if (t == 0) {
    float sum = 0.f;
    int len = 0;
    for (int i = 0; i < 64; ++i) { sum += part[i]; len += lenp[i]; }
    const int first = labels[b * CS];
    const int last  = labels[b * CS + (len - 1)];
    sum += start_t[first] + em[(size_t)b * CS * CT + first] + end_t[last];
    num[b] = sum;
  }
}

// ---------------------------------------------------------------------------
// Final: out = -sum_b (num[b] - den[b]), fixed order (deterministic).
// ---------------------------------------------------------------------------
__global__ __launch_bounds__(32) void crf_final_kernel(
    const float* __restrict__ num, const float* __restrict__ den,
    float* __restrict__ out)
{
  if (threadIdx.x == 0 && blockIdx.x == 0) {
    float s = 0.f;
    for (int b = 0; b < CB; ++b) s += (num[b] - den[b]);
    out[0] = -s;
  }
}

extern "C" void kernel_launch(void* const* d_in, const int* in_sizes, int n_in,
                              void* d_out, int out_size, void* d_ws, size_t ws_size,
                              hipStream_t stream) {
  const float* emissions   = (const float*)d_in[0];
  const int*   attn_mask   = (const int*)d_in[1];
  const int*   labels      = (const int*)d_in[2];
  const float* transitions = (const float*)d_in[3];
  const float* start_t     = (const float*)d_in[4];
  const float* end_t       = (const float*)d_in[5];
  float* out = (float*)d_out;
  float* num = (float*)d_ws;        // 256 floats
  float* den = num + CB;            // 256 floats

  crf_num_kernel<<<CB, 64, 0, stream>>>(emissions, attn_mask, labels,
                                        transitions, start_t, end_t, num);
  crf_den_kernel<<<CB / 16, 32, 0, stream>>>(emissions, attn_mask, transitions,
                                             start_t, end_t, den);
  crf_final_kernel<<<1, 32, 0, stream>>>(num, den, out);
}